// Net_9955734192898
// MI455X (gfx1250) — compile-verified
//
#include <hip/hip_runtime.h>

// ---------------- problem constants ----------------
#define BATCH 64
#define CIN   3
#define HIN   32
#define COUT  2048
#define KSZ   6
#define HOUT  27                 // 32 - 6 + 1
#define NPOS  (HOUT*HOUT)        // 729
#define NTOT  (BATCH*NPOS)       // 46656 = 729 * 64 (no N tail at 64-wide tiles)
#define KDIM  (CIN*KSZ*KSZ)      // 108    (27 WMMA K-steps of 4)
#define KNN   819

typedef __attribute__((ext_vector_type(2))) float v2f;
typedef __attribute__((ext_vector_type(8))) float v8f;

// =====================================================================
// Kernel 1: implicit-GEMM conv via V_WMMA_F32_16X16X4_F32 (full fp32)
//   D[c][p] = sum_k W[c][k] * X[k][p]
//   One wave32 computes a 16(M=cout) x 64(N=pos) strip: 4 accumulators,
//   A fragment reused 4x per K-step. K fully unrolled (27 steps) so all
//   gather offsets constant-fold into load immediates.
// =====================================================================
__global__ __launch_bounds__(256)
void conv_wmma(const float* __restrict__ x,
               const float* __restrict__ w,
               float* __restrict__ out) {
    const int wave  = (blockIdx.x * 256 + threadIdx.x) >> 5;
    const int lane  = threadIdx.x & 31;
    const int nGrp  = NTOT / 64;                 // 729 N-strips
    const int cTile = (wave / nGrp) * 16;
    const int p0    = (wave % nGrp) * 64;
    const int half  = lane >> 4;                 // 0: K=+0,+1  1: K=+2,+3
    const int r     = lane & 15;

    // A rows: one cout per lane-mod-16; k0 = kk+2*half is even -> 8B aligned
    const float* __restrict__ wr = w + (size_t)(cTile + r) * KDIM;

    // decode the 4 output positions this lane's B-columns cover
    const float* xbase[4];
    int bidx[4], rem[4];
#pragma unroll
    for (int t = 0; t < 4; ++t) {
        const int p = p0 + 16 * t + r;
        bidx[t] = p / NPOS;
        rem[t]  = p - bidx[t] * NPOS;
        const int oh = rem[t] / HOUT;
        const int ow = rem[t] - oh * HOUT;
        xbase[t] = x + (size_t)bidx[t] * (CIN * HIN * HIN) + oh * HIN + ow;
    }

    v8f acc[4] = {};
#pragma unroll
    for (int kk = 0; kk < KDIM; kk += 4) {
        // A 16x4 f32 fragment: V0/V1 = K pair for this half-wave
        const v2f a = *(const v2f*)(wr + kk + 2 * half);

        // compile-time image offsets for k = kk .. kk+3
        int off[4];
#pragma unroll
        for (int q = 0; q < 4; ++q) {
            const int k  = kk + q;
            const int ci = k / 36, rr = k - ci * 36;
            const int kh = rr / 6, kw = rr - kh * 6;
            off[q] = ci * HIN * HIN + kh * HIN + kw;   // constant
        }
        const int o_lo = half ? off[2] : off[0];
        const int o_hi = half ? off[3] : off[1];

#pragma unroll
        for (int t = 0; t < 4; ++t) {
            v2f bf;
            bf.x = xbase[t][o_lo];
            bf.y = xbase[t][o_hi];
            acc[t] = __builtin_amdgcn_wmma_f32_16x16x4_f32(
                         false, a, false, bf, (short)0, acc[t], false, false);
        }
    }

    // D 16x16 f32 layout: lane -> N=r, VGPR v -> M = v + 8*half
#pragma unroll
    for (int t = 0; t < 4; ++t) {
        const size_t base = (size_t)bidx[t] * COUT * NPOS + rem[t];
#pragma unroll
        for (int v = 0; v < 8; ++v) {
            const int row = cTile + v + 8 * half;
            out[base + (size_t)row * NPOS] = acc[t][v];
        }
    }
}

// =====================================================================
// Kernel 2: exact rank-819 selection per (b,h,w) over 2048 channels,
//           for both branches (sign = -1 / +1), via 4-pass radix select
//           on order-preserving uint32 keys. One 256-thread block / pos.
// =====================================================================
__device__ __forceinline__ unsigned f2ord(float f) {
    unsigned u = __float_as_uint(f);
    return (u & 0x80000000u) ? ~u : (u | 0x80000000u);
}

__global__ __launch_bounds__(256)
void kth_select(const float* __restrict__ convout,
                const float* __restrict__ bias,
                float* __restrict__ thr) {
    __shared__ float    s_o[COUT];
    __shared__ float    s_b[COUT];
    __shared__ unsigned s_hist[256];
    __shared__ unsigned s_sel[2];   // [digit, remaining rank]

    const int pos  = blockIdx.x;            // 0..NTOT-1
    const int bidx = pos / NPOS;
    const int rem  = pos - bidx * NPOS;
    const float* __restrict__ src =
        convout + (size_t)bidx * COUT * NPOS + rem;

    for (int c = threadIdx.x; c < COUT; c += 256) {
        s_o[c] = src[(size_t)c * NPOS];
        s_b[c] = bias[c];
    }
    __syncthreads();

    for (int br = 0; br < 2; ++br) {
        const float sign = br ? 1.0f : -1.0f;
        unsigned prefix = 0;
        unsigned rank   = KNN;
        for (int pass = 0; pass < 4; ++pass) {
            const int shift = 24 - 8 * pass;
            const unsigned mask =
                (pass == 0) ? 0u : (0xFFFFFFFFu << (shift + 8));
            s_hist[threadIdx.x] = 0;
            __syncthreads();
            for (int c = threadIdx.x; c < COUT; c += 256) {
                const unsigned key = f2ord(sign * s_o[c] + s_b[c]);
                if ((key & mask) == prefix)
                    atomicAdd(&s_hist[(key >> shift) & 0xFFu], 1u);
            }
            __syncthreads();
            if (threadIdx.x == 0) {
                unsigned cum = 0, d = 0, rk = rank;
                for (d = 0; d < 256; ++d) {
                    const unsigned h = s_hist[d];
                    if (cum + h > rank) { rk = rank - cum; break; }
                    cum += h;
                }
                s_sel[0] = d;
                s_sel[1] = rk;
            }
            __syncthreads();
            prefix |= (s_sel[0] << shift);
            rank    = s_sel[1];
        }
        // prefix == orderable key of the (KNN+1)-th smallest; invert
        const unsigned key = prefix;
        const unsigned u   =
            (key & 0x80000000u) ? (key ^ 0x80000000u) : ~key;
        if (threadIdx.x == 0)
            thr[(size_t)br * NTOT + pos] = __uint_as_float(u);
        __syncthreads();
    }
}

// =====================================================================
// Kernel 3: mask = (sign*out + bias < thr), ceil-mode 5x5/s3 avg-pool
//           (valid-count divisors), adaptive 9->6 pool. One block per
//           (b, c), both branches; writes final [2][B][COUT][6][6].
// =====================================================================
__global__ __launch_bounds__(256)
void mask_pool(const float* __restrict__ convout,
               const float* __restrict__ bias,
               const float* __restrict__ thr,
               float* __restrict__ outp) {
    __shared__ float s_o[NPOS];
    __shared__ float s_t[2][NPOS];
    __shared__ float s_p[2][81];

    const int bc   = blockIdx.x;             // b*COUT + c
    const int bidx = bc / COUT;
    const int c    = bc - bidx * COUT;
    const float* __restrict__ src = convout + (size_t)bc * NPOS;

    for (int i = threadIdx.x; i < NPOS; i += 256) {
        s_o[i]    = src[i];
        s_t[0][i] = thr[(size_t)bidx * NPOS + i];
        s_t[1][i] = thr[(size_t)NTOT + bidx * NPOS + i];
    }
    const float bv = bias[c];
    __syncthreads();

    // ceil-mode 5x5 stride-3 avg pool of the 0/1 mask: 27 -> 9
    for (int t = threadIdx.x; t < 162; t += 256) {
        const int br = t / 81;
        const int wi = t - br * 81;
        const int i = wi / 9, j = wi - (wi / 9) * 9;
        const int h0 = i * 3, w0 = j * 3;
        const int hl = min(5, HOUT - h0), wl = min(5, HOUT - w0);
        const float sign = br ? 1.0f : -1.0f;
        float sum = 0.0f;
        for (int dh = 0; dh < hl; ++dh)
            for (int dw = 0; dw < wl; ++dw) {
                const int idx = (h0 + dh) * HOUT + (w0 + dw);
                const float z = sign * s_o[idx] + bv;
                sum += (z < s_t[br][idx]) ? 1.0f : 0.0f;
            }
        s_p[br][wi] = sum / (float)(hl * wl);
    }
    __syncthreads();

    // adaptive avg pool 9 -> 6 (torch bin edges)
    for (int t = threadIdx.x; t < 72; t += 256) {
        const int br = t / 36;
        const int e  = t - br * 36;
        const int i = e / 6, j = e - (e / 6) * 6;
        const int si = (i * 9) / 6, ei = ((i + 1) * 9 + 5) / 6;
        const int sj = (j * 9) / 6, ej = ((j + 1) * 9 + 5) / 6;
        float sum = 0.0f; int cnt = 0;
        for (int a = si; a < ei; ++a)
            for (int b2 = sj; b2 < ej; ++b2) { sum += s_p[br][a * 9 + b2]; ++cnt; }
        outp[(size_t)br * ((size_t)BATCH * COUT * 36) +
             (size_t)bc * 36 + i * 6 + j] = sum / (float)cnt;
    }
}

// =====================================================================
extern "C" void kernel_launch(void* const* d_in, const int* in_sizes, int n_in,
                              void* d_out, int out_size, void* d_ws, size_t ws_size,
                              hipStream_t stream) {
    (void)in_sizes; (void)n_in; (void)out_size; (void)ws_size;
    const float* x    = (const float*)d_in[0];   // [64,3,32,32]
    const float* w    = (const float*)d_in[1];   // [2048,3,6,6]
    const float* bias = (const float*)d_in[2];   // [1,2048,1,1]
    float* outp = (float*)d_out;                 // [2][64][2048][6][6] concat

    // workspace layout: conv out (f32, [B][COUT][27][27]) then thresholds
    float* convout = (float*)d_ws;
    float* thr     = convout + (size_t)BATCH * COUT * NPOS; // [2][B][729]

    // 1) conv: (2048/16) M-tiles * (46656/64) N-strips = 93312 waves
    //    8 waves/block -> 11664 blocks
    conv_wmma<<<(COUT / 16) * (NTOT / 64) / 8, 256, 0, stream>>>(x, w, convout);

    // 2) rank-819 thresholds: one block per spatial position
    kth_select<<<NTOT, 256, 0, stream>>>(convout, bias, thr);

    // 3) mask + pools: one block per (b, c)
    mask_pool<<<BATCH * COUT, 256, 0, stream>>>(convout, bias, thr, outp);
}